// TopologicalSignatureDistance_15307263443421
// MI455X (gfx1250) — compile-verified
//
#include <hip/hip_runtime.h>
#include <stdint.h>

#define NVERT 512
#define NEDGE 131328u       // 512*513/2 upper-tri incl diagonal
#define SORT_N 262144u      // 2^18, padded sort size per matrix

typedef __attribute__((ext_vector_type(2))) float v2f;
typedef __attribute__((ext_vector_type(8))) float v8f;

// ---------- CDNA5 async global->LDS helpers (ASYNCcnt-tracked) ----------
__device__ __forceinline__ void async_load_b128(uint32_t lds_byte_off, const void* gaddr) {
  // GLOBAL_LOAD_ASYNC_TO_LDS_B128: VDST = LDS byte offset VGPR, VADDR = 64-bit global addr
  asm volatile("global_load_async_to_lds_b128 %0, %1, off"
               :: "v"(lds_byte_off), "v"(gaddr) : "memory");
}
__device__ __forceinline__ void wait_asynccnt0() {
  asm volatile("s_wait_asynccnt 0x0" ::: "memory");
}
__device__ __forceinline__ void wait_dscnt0() {
  asm volatile("s_wait_dscnt 0x0" ::: "memory");
}

// ---------- edge index <-> (row,col) for np.triu_indices(512) ----------
__device__ __forceinline__ uint32_t tri_off(uint32_t r) {
  return r * 512u - (r * (r - 1u)) / 2u;   // start index of row r
}
__device__ __forceinline__ void decode_edge(uint32_t e, uint32_t& r, uint32_t& c) {
  float fe = (float)e;
  float disc = 1050625.0f - 8.0f * fe;     // 1025^2 - 8e
  int rr = (int)floorf((1025.0f - sqrtf(disc)) * 0.5f);
  rr = rr < 0 ? 0 : (rr > 511 ? 511 : rr);
  while (rr < 511 && tri_off((uint32_t)(rr + 1)) <= e) ++rr;
  while (rr > 0 && tri_off((uint32_t)rr) > e) --rr;
  r = (uint32_t)rr;
  c = (uint32_t)rr + (e - tri_off((uint32_t)rr));
}

// ---------- kernel 1: build 64-bit sort keys (stable via index in low bits) ----------
__global__ void build_keys_kernel(const float* __restrict__ D1,
                                  const float* __restrict__ D2,
                                  uint64_t* __restrict__ keys) {
  uint32_t t = blockIdx.x * blockDim.x + threadIdx.x;   // 0 .. 2*SORT_N-1
  uint32_t seg = t >> 18;
  uint32_t e = t & (SORT_N - 1u);
  uint64_t key;
  if (e < NEDGE) {
    uint32_t r, c;
    decode_edge(e, r, c);
    const float* D = seg ? D2 : D1;
    float w = D[r * 512u + c];                          // w >= 0: bits monotone
    key = (((uint64_t)__float_as_uint(w)) << 32) | (uint64_t)e;
  } else {
    key = ~0ull;                                        // pad sorts to the end
  }
  keys[t] = key;
}

// ---------- kernel 2: per-512-tile bitonic (stages k=2..512) in LDS ----------
__global__ void tile_sort_kernel(uint64_t* __restrict__ keys) {
  __shared__ uint64_t tile[512];
  uint32_t tid = threadIdx.x;                           // 0..255
  uint64_t base = (uint64_t)blockIdx.x * 512u;
  uint32_t lbase = (uint32_t)(base & (SORT_N - 1u));    // segment-local base

  uint32_t loff = (uint32_t)(uintptr_t)(&tile[0]) + tid * 16u;
  async_load_b128(loff, keys + base + (uint64_t)tid * 2u);
  wait_asynccnt0();
  __syncthreads();

  for (uint32_t k = 2; k <= 512u; k <<= 1) {
    for (uint32_t j = k >> 1; j >= 1u; j >>= 1) {
      uint32_t i = ((tid & ~(j - 1u)) << 1) | (tid & (j - 1u));  // bit j clear
      uint32_t p = i + j;
      bool asc = (((lbase + i) & k) == 0u);
      uint64_t a = tile[i], b = tile[p];
      if (asc ? (a > b) : (a < b)) { tile[i] = b; tile[p] = a; }
      __syncthreads();
    }
  }
  keys[base + tid * 2u]      = tile[tid * 2u];
  keys[base + tid * 2u + 1u] = tile[tid * 2u + 1u];
}

// ---------- kernel 3: global compare-exchange stage (j >= 512) ----------
__global__ void global_stage_kernel(uint64_t* __restrict__ keys, uint32_t k, uint32_t j) {
  uint32_t t = blockIdx.x * blockDim.x + threadIdx.x;   // 0 .. 2*(SORT_N/2)-1
  uint32_t seg = t >> 17;                               // SORT_N/2 = 2^17
  uint32_t p = t & ((SORT_N >> 1) - 1u);
  uint32_t i = ((p & ~(j - 1u)) << 1) | (p & (j - 1u));
  uint32_t q = i + j;
  bool asc = ((i & k) == 0u);
  uint64_t* base = keys + (uint64_t)seg * SORT_N;
  uint64_t a = base[i], b = base[q];
  if (asc ? (a > b) : (a < b)) { base[i] = b; base[q] = a; }
}

// ---------- kernel 4: fused tile merge (stages j=256..1) for a given k ----------
__global__ void tile_merge_kernel(uint64_t* __restrict__ keys, uint32_t k) {
  __shared__ uint64_t tile[512];
  uint32_t tid = threadIdx.x;
  uint64_t base = (uint64_t)blockIdx.x * 512u;
  uint32_t lbase = (uint32_t)(base & (SORT_N - 1u));
  bool asc = ((lbase & k) == 0u);                       // uniform: k >= 1024 > tile span

  uint32_t loff = (uint32_t)(uintptr_t)(&tile[0]) + tid * 16u;
  async_load_b128(loff, keys + base + (uint64_t)tid * 2u);
  wait_asynccnt0();
  __syncthreads();

  for (uint32_t j = 256u; j >= 1u; j >>= 1) {
    uint32_t i = ((tid & ~(j - 1u)) << 1) | (tid & (j - 1u));
    uint32_t p = i + j;
    uint64_t a = tile[i], b = tile[p];
    if (asc ? (a > b) : (a < b)) { tile[i] = b; tile[p] = a; }
    __syncthreads();
  }
  keys[base + tid * 2u]      = tile[tid * 2u];
  keys[base + tid * 2u + 1u] = tile[tid * 2u + 1u];
}

// ---------- kernel 5: Kruskal union-find (1 wave32 per matrix, parent[] in LDS) ----------
__global__ void kruskal_kernel(const uint64_t* __restrict__ keys, int* __restrict__ pairs) {
  __shared__ int parent[NVERT];
  __shared__ uint64_t buf[512];
  __shared__ int s_count;

  int seg = blockIdx.x;
  const uint64_t* skeys = keys + (uint64_t)seg * SORT_N;
  int* pu = pairs + seg * 1024;   // 511 used
  int* pv = pu + 512;

  for (int i = (int)threadIdx.x; i < NVERT; i += 32) parent[i] = i;
  if (threadIdx.x == 0) s_count = 0;
  __syncthreads();

  uint32_t lbuf = (uint32_t)(uintptr_t)(&buf[0]);

  for (uint32_t chunk = 0; chunk < SORT_N; chunk += 512u) {
    wait_dscnt0();   // ensure prior-chunk LDS reads retired before async engine rewrites buf
    for (uint32_t rnd = 0; rnd < 8u; ++rnd) {
      uint32_t slot = rnd * 32u + threadIdx.x;          // 0..255, 16B each
      async_load_b128(lbuf + slot * 16u, skeys + chunk + (uint64_t)slot * 2u);
    }
    wait_asynccnt0();
    __syncthreads();

    if (threadIdx.x == 0) {
      int count = s_count;
      for (int m = 0; m < 512 && count < NVERT - 1; ++m) {
        uint64_t kk = buf[m];
        uint32_t e = (uint32_t)kk;
        if (e >= NEDGE) continue;
        uint32_t r, c;
        decode_edge(e, r, c);
        if (r == c) continue;                            // diagonal: never merges
        int ru = (int)r;                                 // path-halving find
        while (parent[ru] != ru) { parent[ru] = parent[parent[ru]]; ru = parent[ru]; }
        int rv = (int)c;
        while (parent[rv] != rv) { parent[rv] = parent[parent[rv]]; rv = parent[rv]; }
        if (ru != rv) {
          int lo = ru < rv ? ru : rv;
          int hi = ru < rv ? rv : ru;
          parent[lo] = hi;                               // matches reference merge rule
          pu[count] = (int)r;                            // r < c always here
          pv[count] = (int)c;
          ++count;
        }
      }
      s_count = count;
    }
    __syncthreads();
    if (s_count >= NVERT - 1) break;                     // uniform exit
  }
}

// ---------- kernel 6: gather signatures + WMMA squared-diff reduction ----------
// One full wave32 (EXEC all ones, required by WMMA). Squared differences are
// staged in zero-padded LDS, then reduced with V_WMMA_F32_16X16X4_F32 using
// B = all-ones: D[m][n] = sum_k A[m][k], so sum(all D) = 16 * sum(all A slots)
// regardless of the A/D lane<->element mapping. Multiply by 1/16 (exact).
__global__ void reduce_kernel(const float* __restrict__ D1, const float* __restrict__ D2,
                              const int* __restrict__ pairs, float* __restrict__ out) {
  __shared__ float diffs[1024];   // [0,512): set1 diffs, [512,1024): set2 diffs, padded 0
  __shared__ float acc[256];
  int t = (int)threadIdx.x;       // 0..31

  for (int g = t; g < 1024; g += 32) {
    float v = 0.0f;
    if (g < NVERT - 1) {
      int u = pairs[g], w = pairs[512 + g];
      float d = D1[u * 512 + w] - D2[u * 512 + w];
      v = d * d;
    } else if (g >= 512 && g < 512 + NVERT - 1) {
      int idx = g - 512;
      int u = pairs[1024 + idx], w = pairs[1536 + idx];
      float d = D2[u * 512 + w] - D1[u * 512 + w];
      v = d * d;
    }
    diffs[g] = v;
  }
  __syncthreads();

  v8f c = {};
  v2f b;
  b.x = 1.0f; b.y = 1.0f;
  for (int it = 0; it < 16; ++it) {        // uniform loop, EXEC all ones
    v2f a;
    a.x = diffs[it * 64 + t * 2];
    a.y = diffs[it * 64 + t * 2 + 1];
    c = __builtin_amdgcn_wmma_f32_16x16x4_f32(false, a, false, b,
                                              (short)0, c, false, false);
  }

  for (int i = 0; i < 8; ++i) acc[t * 8 + i] = c[i];
  __syncthreads();
  if (t == 0) {
    float s = 0.0f;
    for (int i = 0; i < 256; ++i) s += acc[i];
    out[0] = s * 0.0625f;                  // undo 16x column replication (exact)
  }
}

// ---------- host-side launch sequence (graph-capture safe, fixed schedule) ----------
extern "C" void kernel_launch(void* const* d_in, const int* in_sizes, int n_in,
                              void* d_out, int out_size, void* d_ws, size_t ws_size,
                              hipStream_t stream) {
  const float* D1 = (const float*)d_in[0];
  const float* D2 = (const float*)d_in[1];
  float* out = (float*)d_out;

  const size_t keys_bytes = (size_t)2 * SORT_N * sizeof(uint64_t);   // 4 MB
  if (ws_size < keys_bytes + 4096 * sizeof(int)) return;
  uint64_t* keys = (uint64_t*)d_ws;
  int* pairs = (int*)((char*)d_ws + keys_bytes);                     // [2][2][512]

  build_keys_kernel<<<(2 * SORT_N) / 256, 256, 0, stream>>>(D1, D2, keys);

  // bitonic: tile-local k=2..512, then per-k global stages (j>=512) + fused LDS tail
  tile_sort_kernel<<<(2 * SORT_N) / 512, 256, 0, stream>>>(keys);
  for (uint32_t k = 1024; k <= SORT_N; k <<= 1) {
    for (uint32_t j = k >> 1; j >= 512u; j >>= 1)
      global_stage_kernel<<<SORT_N / 256, 256, 0, stream>>>(keys, k, j);
    tile_merge_kernel<<<(2 * SORT_N) / 512, 256, 0, stream>>>(keys, k);
  }

  kruskal_kernel<<<2, 32, 0, stream>>>(keys, pairs);
  reduce_kernel<<<1, 32, 0, stream>>>(D1, D2, pairs, out);
}